// TDrumorGCN_20194936226502
// MI455X (gfx1250) — compile-verified
//
#include <hip/hip_runtime.h>

typedef __attribute__((ext_vector_type(2))) float v2f;
typedef __attribute__((ext_vector_type(8))) float v8f;

#define NODES 100000
#define EDGES 1600000
#define IN_F  512
#define HID   128
#define OUT_F 256
#define NGRAPH 128

// ---------------------------------------------------------------- utilities
__global__ void fill_kernel(float* __restrict__ p, float v, int n) {
    int i = blockIdx.x * blockDim.x + threadIdx.x;
    if (i < n) p[i] = v;
}

__global__ void deg_accum_kernel(const int* __restrict__ dst,
                                 float* __restrict__ deg, int e) {
    int i = blockIdx.x * blockDim.x + threadIdx.x;
    if (i < e) atomicAdd(&deg[dst[i]], 1.0f);
}

__global__ void rsqrt_kernel(float* __restrict__ deg, int n) {
    int i = blockIdx.x * blockDim.x + threadIdx.x;
    if (i < n) deg[i] = rsqrtf(deg[i]);
}

__global__ void bias_relu_kernel(float* __restrict__ h,
                                 const float* __restrict__ b,
                                 int F, int total) {
    int i = blockIdx.x * blockDim.x + threadIdx.x;
    if (i < total) {
        int c = i & (F - 1);
        h[i] = fmaxf(h[i] + b[c], 0.0f);
    }
}

__global__ void init_bias_kernel(float* __restrict__ h,
                                 const float* __restrict__ b,
                                 int F, int total) {
    int i = blockIdx.x * blockDim.x + threadIdx.x;
    if (i < total) h[i] = b[i & (F - 1)];
}

// ------------------------------------------------- WMMA f32 GEMM: C = A * W
// A: [nrows x K] row-major (nrows multiple of 16), W: [K x Nout] row-major.
// Block: 256 threads = 8 waves. Block tile: 16 rows x 128 cols (wave = 16x16).
// The 16xK A tile is staged in LDS via CDNA5 async loads
// (global_load_async_to_lds_b128, ASYNCcnt-tracked) with a +4 float row pad:
// row stride (K+4)*4 bytes is 16B-aligned (required for B128 LDS writes) and
// rotates LDS banks by 4 dwords per row so the 16 lanes of each ds_load_b64
// A-fragment read hit distinct banks.
// Fragments follow the CDNA5 ISA 16x4 f32 A layout:
//   lane L: M = L%16, VGPR0 holds K = 2*(L>>4), VGPR1 holds K = 2*(L>>4)+1.
__global__ void gemm_wmma_f32_kernel(const float* __restrict__ A,
                                     const float* __restrict__ W,
                                     float* __restrict__ C,
                                     int K, int Nout) {
    extern __shared__ float smem[];            // 16 * (K+4) floats
    const int ldS = K + 4;
    const int m0  = blockIdx.x * 16;

    // async cooperative load of the 16xK A tile (16B per lane per issue;
    // chunks never cross a row since K % 4 == 0)
    for (int i = threadIdx.x * 4; i < 16 * K; i += 256 * 4) {
        const int r = i / K;
        const int c = i - r * K;
        const unsigned lds_addr = (unsigned)(size_t)&smem[r * ldS + c];
        const float*   gaddr    = A + (size_t)(m0 + r) * K + c;
        asm volatile("global_load_async_to_lds_b128 %0, %1, off"
                     :: "v"(lds_addr), "v"(gaddr) : "memory");
    }
    asm volatile("s_wait_asynccnt 0x0" ::: "memory");
    __syncthreads();

    const int wave  = threadIdx.x >> 5;        // 0..7
    const int lane  = threadIdx.x & 31;
    const int n0    = blockIdx.y * 128 + wave * 16;
    const int mrow  = lane & 15;               // M within tile
    const int khalf = (lane >> 4) << 1;        // 0 or 2
    const int col   = n0 + (lane & 15);

    v8f acc = {};
    const float2* arow =
        (const float2*)(smem + mrow * ldS);    // ldS even -> 8B aligned

    #pragma unroll 4
    for (int k = 0; k < K; k += 4) {
        const int kb = k + khalf;
        float2 a2 = arow[kb >> 1];             // ds_load_b64
        v2f af = { a2.x, a2.y };
        v2f bf = { W[(size_t)kb * Nout + col],
                   W[(size_t)(kb + 1) * Nout + col] };
        acc = __builtin_amdgcn_wmma_f32_16x16x4_f32(
            false, af, false, bf, (short)0, acc, false, false);
    }

    // C/D layout: VGPR r -> row m0 + r + 8*(lane>=16), col n0 + (lane&15)
    const int rbase = m0 + ((lane >> 4) << 3);
    #pragma unroll
    for (int r = 0; r < 8; ++r) {
        C[(size_t)(rbase + r) * Nout + col] = acc[r];
    }
}

// ------------------------------------------ edge-parallel scatter (GCN norm)
// F threads per edge (F = 128 or 256); edges [0,E) from arrays, [E,total) are
// self-loops (src = dst = e - E).
__global__ void scatter_edges_kernel(const float* __restrict__ hsrc,
                                     const int* __restrict__ srcIdx,
                                     const int* __restrict__ dstIdx,
                                     const float* __restrict__ dinv,
                                     float* __restrict__ out,
                                     int F, int E, int total) {
    const int per_block = blockDim.x / F;
    const int e = blockIdx.x * per_block + (int)(threadIdx.x / F);
    const int c = threadIdx.x & (F - 1);
    if (e >= total) return;
    int s, d;
    if (e < E) { s = srcIdx[e]; d = dstIdx[e]; }
    else       { s = e - E; d = s; }
    const float norm = dinv[s] * dinv[d];
    const float val  = hsrc[(size_t)s * F + c] * norm;
    atomicAdd(&out[(size_t)d * F + c], val);
}

// ------------------------------------------------------- global add pooling
__global__ void pool_kernel(const float* __restrict__ h,
                            const int* __restrict__ batch,
                            float* __restrict__ hs,
                            int F, int total) {
    int i = blockIdx.x * blockDim.x + threadIdx.x;
    if (i < total) {
        int n = i / F;
        int c = i - n * F;
        atomicAdd(&hs[(size_t)batch[n] * F + c], h[i]);
    }
}

// ---------------------------------------------------------------- launcher
extern "C" void kernel_launch(void* const* d_in, const int* in_sizes, int n_in,
                              void* d_out, int out_size, void* d_ws, size_t ws_size,
                              hipStream_t stream) {
    const float* x    = (const float*)d_in[0];            // [N, 512]
    const int*   ei   = (const int*)  d_in[1];            // [2, E]
    const int*   bat  = (const int*)  d_in[2];            // [N]
    const float* W1   = (const float*)d_in[3];            // [512, 128]
    const float* b1   = (const float*)d_in[4];            // [128]
    const float* W2   = (const float*)d_in[5];            // [128, 256]
    const float* b2   = (const float*)d_in[6];            // [256]

    const int* src = ei;
    const int* dst = ei + EDGES;

    float* hs   = (float*)d_out;                          // [128, 256]
    float* hout = hs + (size_t)NGRAPH * OUT_F;            // [N, 256]

    float* ws   = (float*)d_ws;
    float* deg  = ws;                                     // N      (-> dinv)
    float* xw1  = deg + NODES;                            // N*128
    float* h1   = xw1 + (size_t)NODES * HID;              // N*128
    float* xw2  = h1  + (size_t)NODES * HID;              // N*256

    const int total_e = EDGES + NODES;                    // with self-loops

    // degree / normalization
    fill_kernel<<<(NODES + 255) / 256, 256, 0, stream>>>(deg, 1.0f, NODES);
    deg_accum_kernel<<<(EDGES + 255) / 256, 256, 0, stream>>>(dst, deg, EDGES);
    rsqrt_kernel<<<(NODES + 255) / 256, 256, 0, stream>>>(deg, NODES);

    // layer 1: xw1 = x @ W1   (K=512, Nout=128)
    {
        dim3 grid(NODES / 16, HID / 128);
        size_t smem = (size_t)16 * (IN_F + 4) * sizeof(float);
        gemm_wmma_f32_kernel<<<grid, 256, smem, stream>>>(x, W1, xw1, IN_F, HID);
    }
    fill_kernel<<<((size_t)NODES * HID + 255) / 256, 256, 0, stream>>>(
        h1, 0.0f, NODES * HID);
    scatter_edges_kernel<<<(total_e + 1) / 2, 256, 0, stream>>>(
        xw1, src, dst, deg, h1, HID, EDGES, total_e);
    bias_relu_kernel<<<((size_t)NODES * HID + 255) / 256, 256, 0, stream>>>(
        h1, b1, HID, NODES * HID);

    // layer 2: xw2 = h1 @ W2   (K=128, Nout=256)
    {
        dim3 grid(NODES / 16, OUT_F / 128);
        size_t smem = (size_t)16 * (HID + 4) * sizeof(float);
        gemm_wmma_f32_kernel<<<grid, 256, smem, stream>>>(h1, W2, xw2, HID, OUT_F);
    }
    init_bias_kernel<<<((size_t)NODES * OUT_F + 255) / 256, 256, 0, stream>>>(
        hout, b2, OUT_F, NODES * OUT_F);
    scatter_edges_kernel<<<total_e, 256, 0, stream>>>(
        xw2, src, dst, deg, hout, OUT_F, EDGES, total_e);

    // pooling
    fill_kernel<<<(NGRAPH * OUT_F + 255) / 256, 256, 0, stream>>>(
        hs, 0.0f, NGRAPH * OUT_F);
    pool_kernel<<<((size_t)NODES * OUT_F + 255) / 256, 256, 0, stream>>>(
        hout, bat, hs, OUT_F, NODES * OUT_F);
}